// GraphModel_73607149519132
// MI455X (gfx1250) — compile-verified
//
#include <hip/hip_runtime.h>

typedef float v2f __attribute__((ext_vector_type(2)));
typedef float v8f __attribute__((ext_vector_type(8)));
typedef unsigned int u32x4 __attribute__((ext_vector_type(4)));
typedef int i32x4 __attribute__((ext_vector_type(4)));
typedef int i32x8 __attribute__((ext_vector_type(8)));

#define HDIM 128
#define LROW 132            // padded LDS row stride (dwords): conflict-free WMMA reads
#define LNEPS 1e-5f

// ---------------------------------------------------------------------------
// x[n] = key_emb[tok0[n]] + val_emb[tok1[n]] ; deg[n] = 1 (self loop)
// ---------------------------------------------------------------------------
__global__ void k_embed(const int* __restrict__ tok,
                        const float* __restrict__ kemb,
                        const float* __restrict__ vemb,
                        float* __restrict__ x,
                        float* __restrict__ deg, int N) {
    int i = blockIdx.x * blockDim.x + threadIdx.x;
    int n = i >> 5;
    if (n >= N) return;
    int h = (i & 31) << 2;
    int t0 = tok[2 * n + 0];
    int t1 = tok[2 * n + 1];
    float4 a = *(const float4*)(kemb + (size_t)t0 * HDIM + h);
    float4 b = *(const float4*)(vemb + (size_t)t1 * HDIM + h);
    float4 r = make_float4(a.x + b.x, a.y + b.y, a.z + b.z, a.w + b.w);
    *(float4*)(x + (size_t)n * HDIM + h) = r;
    if ((i & 31) == 0) deg[n] = 1.0f;
}

__global__ void k_deg(const int* __restrict__ dstIdx, float* __restrict__ deg, int E) {
    int e = blockIdx.x * blockDim.x + threadIdx.x;
    if (e < E) unsafeAtomicAdd(&deg[dstIdx[e]], 1.0f);
}

__global__ void k_dis(const float* __restrict__ deg, float* __restrict__ dis, int N) {
    int n = blockIdx.x * blockDim.x + threadIdx.x;
    if (n < N) dis[n] = rsqrtf(deg[n]);   // deg >= 1 always (self loop)
}

// ---------------------------------------------------------------------------
// xw = x @ W   (N x 128)@(128 x 128), fp32 WMMA 16x16x4.
// Block = 8 waves = 128 rows.
//  - x panel (128 x 128 f32) loaded by the Tensor Data Mover into LDS with
//    pad_enable (4 dwords every 128) -> 132-dword rows, bank-conflict-free
//    A-operand ds_load_b64; hardware zero-fills rows past N (tensor_dim1).
//  - W staged transposed (WT[n][k], 132-dword rows) so the B-operand pair
//    {W[k][n], W[k+1][n]} is one contiguous ds_load_b64.
// Inner loop: 9 ds_load_b64 + 8 v_wmma per K-step, no register shuffling.
// ---------------------------------------------------------------------------
__global__ void __launch_bounds__(256) k_gemm_xw(const float* __restrict__ x,
                                                 const float* __restrict__ W,
                                                 float* __restrict__ xw, int N) {
    __shared__ float lx[HDIM * LROW];    // 66 KB: x panel, padded rows
    __shared__ float lwt[HDIM * LROW];   // 66 KB: W transposed, padded rows

    int row0blk = blockIdx.x * 128;

    if (threadIdx.x < 32) {              // wave 0 issues the async TDM copy
        unsigned long long ga =
            (unsigned long long)(const void*)(x + (size_t)row0blk * HDIM);
        unsigned lds = (unsigned)(unsigned long long)(const void*)lx;
        unsigned rem = (unsigned)(N - row0blk);       // valid rows; rest read 0
        u32x4 g0;
        g0.x = 1u;                                    // count=1, user mode
        g0.y = lds;                                   // lds_addr
        g0.z = (unsigned)ga;                          // global_addr[31:0]
        g0.w = ((unsigned)(ga >> 32) & 0x01FFFFFFu)   // global_addr[56:32]
               | 0x80000000u;                         // type=2 ("image")
        i32x8 g1;
        g1[0] = 0x07920000;        // data_size=4B, pad_en, interval=128dw, pad=4dw
        g1[1] = (int)(128u << 16); // tensor_dim0 = 128
        g1[2] = (int)((rem & 0xFFFFu) << 16);             // tensor_dim1 lo16
        g1[3] = (int)((rem >> 16) | (128u << 16));        // dim1 hi | tile_dim0=128
        g1[4] = 128;               // tile_dim1 = 128 rows, tile_dim2 = 0
        g1[5] = 128;               // tensor_dim0_stride = 128
        g1[6] = 0;
        g1[7] = 0;
        i32x4 z4 = {0, 0, 0, 0};
        i32x8 z8 = {0, 0, 0, 0, 0, 0, 0, 0};
        __builtin_amdgcn_tensor_load_to_lds(g0, g1, z4, z4, z8, 0);
    }

    // All 256 threads stage W^T (coalesced float4 global reads, LDS scatter)
    for (int i = 0; i < 16; ++i) {
        int e4 = threadIdx.x + i * 256;          // 4096 float4 chunks of W
        int k  = e4 >> 5;                        // 32 float4 per 128-wide row
        int n4 = (e4 & 31) << 2;
        float4 v = *(const float4*)(W + (size_t)k * HDIM + n4);
        lwt[(n4 + 0) * LROW + k] = v.x;
        lwt[(n4 + 1) * LROW + k] = v.y;
        lwt[(n4 + 2) * LROW + k] = v.z;
        lwt[(n4 + 3) * LROW + k] = v.w;
    }
    if (threadIdx.x < 32) __builtin_amdgcn_s_wait_tensorcnt(0);
    __syncthreads();

    int nTiles = (N + 15) >> 4;
    int tile = blockIdx.x * 8 + (threadIdx.x >> 5);
    if (tile < nTiles) {
        int lane = threadIdx.x & 31;
        int m  = lane & 15;
        int kh = lane >> 4;
        int lr = ((threadIdx.x >> 5) << 4) + m;            // local row in panel
        const float* apl = lx  + (size_t)lr * LROW + (kh << 1);
        const float* bpl = lwt + (size_t)m  * LROW + (kh << 1);

        v8f c[8] = {};
        for (int k0 = 0; k0 < HDIM; k0 += 4) {
            v2f a = *(const v2f*)(apl + k0);
#pragma unroll
            for (int t = 0; t < 8; ++t) {
                v2f b = *(const v2f*)(bpl + t * (16 * LROW) + k0);
                c[t] = __builtin_amdgcn_wmma_f32_16x16x4_f32(
                    false, a, false, b, (short)0, c[t], false, false);
            }
        }
        int row0 = tile << 4;
        int rbase = row0 + (kh << 3);
#pragma unroll
        for (int j = 0; j < 8; ++j) {
            int row = rbase + j;
            if (row < N) {
                float* op = xw + (size_t)row * HDIM + m;
#pragma unroll
                for (int t = 0; t < 8; ++t) op[t * 16] = c[t][j];
            }
        }
    }
}

// ---------------------------------------------------------------------------
// logits = x @ out_W^T   (N x 128)@(128 x 32); out_W is [32][128] row-major
// B pairs {out_W[o][k], out_W[o][k+1]} are contiguous b64 global loads.
// ---------------------------------------------------------------------------
__global__ void k_gemm_out(const float* __restrict__ x, const float* __restrict__ outW,
                           float* __restrict__ logits, int N) {
    int wid  = (blockIdx.x * blockDim.x + threadIdx.x) >> 5;
    int lane = threadIdx.x & 31;
    int nTiles = (N + 15) >> 4;
    if (wid >= nTiles) return;
    int row0 = wid << 4;
    int m  = lane & 15;
    int kh = lane >> 4;
    int arow = row0 + m;
    if (arow >= N) arow = N - 1;
    const float* ap = x + (size_t)arow * HDIM + (kh << 1);

    v8f c[2] = {};
    for (int k0 = 0; k0 < HDIM; k0 += 4) {
        v2f a = *(const v2f*)(ap + k0);
#pragma unroll
        for (int t = 0; t < 2; ++t) {
            v2f b = *(const v2f*)(outW + (size_t)(t * 16 + m) * HDIM + k0 + (kh << 1));
            c[t] = __builtin_amdgcn_wmma_f32_16x16x4_f32(
                false, a, false, b, (short)0, c[t], false, false);
        }
    }
    int rbase = row0 + (kh << 3);
#pragma unroll
    for (int j = 0; j < 8; ++j) {
        int row = rbase + j;
        if (row < N) {
#pragma unroll
            for (int t = 0; t < 2; ++t)
                logits[(size_t)row * 32 + t * 16 + m] = c[t][j];
        }
    }
}

// ---------------------------------------------------------------------------
// agg[n] = (1/deg[n]) * xw[n] + bias        (self-loop term + bias)
// ---------------------------------------------------------------------------
__global__ void k_agg_init(const float* __restrict__ xw, const float* __restrict__ dis,
                           const float* __restrict__ bias, float* __restrict__ agg, int N) {
    int i = blockIdx.x * blockDim.x + threadIdx.x;
    int n = i >> 5;
    if (n >= N) return;
    int h = (i & 31) << 2;
    float sn = dis[n];
    sn *= sn;
    float4 v = *(const float4*)(xw + (size_t)n * HDIM + h);
    float4 b = *(const float4*)(bias + h);
    float4 r = make_float4(fmaf(sn, v.x, b.x), fmaf(sn, v.y, b.y),
                           fmaf(sn, v.z, b.z), fmaf(sn, v.w, b.w));
    *(float4*)(agg + (size_t)n * HDIM + h) = r;
}

// ---------------------------------------------------------------------------
// agg[dst[e]] += dis[src]*dis[dst] * xw[src[e]]   (32 lanes/edge, f32 atomics)
// ---------------------------------------------------------------------------
__global__ void k_scatter(const int* __restrict__ srcIdx, const int* __restrict__ dstIdx,
                          const float* __restrict__ dis, const float* __restrict__ xw,
                          float* __restrict__ agg, int E) {
    long long i = (long long)blockIdx.x * blockDim.x + threadIdx.x;
    int e = (int)(i >> 5);
    if (e >= E) return;
    int h = ((int)i & 31) << 2;
    int s = srcIdx[e];
    int d = dstIdx[e];
    float w = dis[s] * dis[d];
    float4 v = *(const float4*)(xw + (size_t)s * HDIM + h);
    float* op = agg + (size_t)d * HDIM + h;
    unsafeAtomicAdd(op + 0, w * v.x);
    unsafeAtomicAdd(op + 1, w * v.y);
    unsafeAtomicAdd(op + 2, w * v.z);
    unsafeAtomicAdd(op + 3, w * v.w);
}

// ---------------------------------------------------------------------------
// x = layernorm(x + relu(agg)) * g + b   — one wave per 128-wide row
// ---------------------------------------------------------------------------
__global__ void k_ln(float* __restrict__ x, const float* __restrict__ agg,
                     const float* __restrict__ g, const float* __restrict__ b, int N) {
    int i = blockIdx.x * blockDim.x + threadIdx.x;
    int n = i >> 5;
    if (n >= N) return;
    int h = (i & 31) << 2;
    float4 a  = *(const float4*)(agg + (size_t)n * HDIM + h);
    float4 xv = *(const float4*)(x   + (size_t)n * HDIM + h);
    float4 y;
    y.x = xv.x + fmaxf(a.x, 0.f);
    y.y = xv.y + fmaxf(a.y, 0.f);
    y.z = xv.z + fmaxf(a.z, 0.f);
    y.w = xv.w + fmaxf(a.w, 0.f);
    float s = y.x + y.y + y.z + y.w;
#pragma unroll
    for (int off = 16; off > 0; off >>= 1) s += __shfl_xor(s, off, 32);
    float mu = s * (1.0f / HDIM);
    float dx = y.x - mu, dy = y.y - mu, dz = y.z - mu, dw = y.w - mu;
    float q = dx * dx + dy * dy + dz * dz + dw * dw;
#pragma unroll
    for (int off = 16; off > 0; off >>= 1) q += __shfl_xor(q, off, 32);
    float r = rsqrtf(q * (1.0f / HDIM) + LNEPS);
    float4 gv = *(const float4*)(g + h);
    float4 bv = *(const float4*)(b + h);
    float4 o = make_float4(dx * r * gv.x + bv.x, dy * r * gv.y + bv.y,
                           dz * r * gv.z + bv.z, dw * r * gv.w + bv.w);
    *(float4*)(x + (size_t)n * HDIM + h) = o;
}

// ---------------------------------------------------------------------------
extern "C" void kernel_launch(void* const* d_in, const int* in_sizes, int n_in,
                              void* d_out, int out_size, void* d_ws, size_t ws_size,
                              hipStream_t stream) {
    const int*   x_tokens   = (const int*)d_in[0];
    const int*   edge_index = (const int*)d_in[1];
    // d_in[2] = batch (unused by the reference output)
    const float* key_emb = (const float*)d_in[3];
    const float* val_emb = (const float*)d_in[4];
    const float* Ws      = (const float*)d_in[5];
    const float* bs      = (const float*)d_in[6];
    const float* ln_g    = (const float*)d_in[7];
    const float* ln_b    = (const float*)d_in[8];
    const float* out_W   = (const float*)d_in[9];
    float* logits = (float*)d_out;

    int N = in_sizes[0] / 2;
    int E = in_sizes[1] / 2;
    int L = in_sizes[5] / (HDIM * HDIM);

    char* p = (char*)d_ws;
    float* x   = (float*)p;  p += (size_t)N * HDIM * sizeof(float);
    float* xw  = (float*)p;  p += (size_t)N * HDIM * sizeof(float);
    float* agg = (float*)p;  p += (size_t)N * HDIM * sizeof(float);
    float* deg = (float*)p;  p += (size_t)N * sizeof(float);
    float* dis = (float*)p;

    const int* src = edge_index;
    const int* dst = edge_index + E;

    const int TH = 256;
    int nodeBlocks  = (N * 32 + TH - 1) / TH;                 // 32 threads/node
    int nTiles      = (N + 15) / 16;
    int gemmBlocks  = (nTiles + 7) / 8;                       // 8 waves (tiles)/block
    int outBlocks   = (nTiles * 32 + TH - 1) / TH;
    long long scT   = (long long)E * 32;
    int scatBlocks  = (int)((scT + TH - 1) / TH);

    k_embed<<<nodeBlocks, TH, 0, stream>>>(x_tokens, key_emb, val_emb, x, deg, N);
    k_deg<<<(E + TH - 1) / TH, TH, 0, stream>>>(dst, deg, E);
    k_dis<<<(N + TH - 1) / TH, TH, 0, stream>>>(deg, dis, N);

    for (int i = 0; i < L; ++i) {
        k_gemm_xw<<<gemmBlocks, TH, 0, stream>>>(x, Ws + (size_t)i * HDIM * HDIM, xw, N);
        k_agg_init<<<nodeBlocks, TH, 0, stream>>>(xw, dis, bs + (size_t)i * HDIM, agg, N);
        k_scatter<<<scatBlocks, TH, 0, stream>>>(src, dst, dis, xw, agg, E);
        k_ln<<<nodeBlocks, TH, 0, stream>>>(x, agg, ln_g + (size_t)i * HDIM,
                                            ln_b + (size_t)i * HDIM, N);
    }
    k_gemm_out<<<outBlocks, TH, 0, stream>>>(x, out_W, logits, N);
}